// GraphLayer_66967130079530
// MI455X (gfx1250) — compile-verified
//
#include <hip/hip_runtime.h>
#include <hip/hip_bf16.h>
#include <stdint.h>

// ---------------------------------------------------------------------------
// GraphLayer: deg-histogram -> edge weights -> scatter-add aggregate -> GEMM.
// MI455X reasoning: scatter phase is HBM-bound (~2 GB traffic @ 23.3 TB/s);
// GEMM phase (3.3 GFLOP fp32) uses V_WMMA_F32_16X16X4_F32 so the matrix pipe
// does the dense part. All fp32 end-to-end to match the fp32 reference.
// ---------------------------------------------------------------------------

typedef __attribute__((ext_vector_type(2))) float v2f;
typedef __attribute__((ext_vector_type(8))) float v8f;

#define D_FEAT 128

// ---- kernel 1: zero deg + x accumulator (float4 stores) --------------------
__global__ void zero_f4_kernel(float4* __restrict__ p, int n4) {
    int i = blockIdx.x * blockDim.x + threadIdx.x;
    if (i < n4) p[i] = make_float4(0.f, 0.f, 0.f, 0.f);
}

// ---- kernel 2: degree histogram over source nodes --------------------------
__global__ void deg_kernel(const int* __restrict__ n1, float* __restrict__ deg, int E) {
    int e = blockIdx.x * blockDim.x + threadIdx.x;
    if (e < E) atomicAdd(&deg[n1[e]], 1.0f);
}

// ---- kernel 3: per-edge weight w = rsqrt(deg1*deg2) * exp(-d^2) ------------
__global__ void weight_kernel(const int* __restrict__ n1, const int* __restrict__ n2,
                              const float* __restrict__ dist, const float* __restrict__ deg,
                              float* __restrict__ w, int E) {
    int e = blockIdx.x * blockDim.x + threadIdx.x;
    if (e >= E) return;
    float d1 = deg[n1[e]];
    float d2 = deg[n2[e]];
    float d  = dist[e];
    w[e] = rsqrtf(d1 * d2) * __expf(-d * d);
}

// ---- kernel 4: scatter-add  x[n1] += w * poi[n2]  (one wave32 per edge) ----
// Each lane handles 4 contiguous floats: float4 gather + 4 f32 global atomics.
__global__ void __launch_bounds__(256)
scatter_kernel(const float* __restrict__ poi, const int* __restrict__ n1s,
               const int* __restrict__ n2s, const float* __restrict__ w,
               float* __restrict__ x, int E) {
    int gid  = blockIdx.x * blockDim.x + threadIdx.x;
    int e    = gid >> 5;          // edge per wave32
    int lane = gid & 31;
    if (e >= E) return;
    int   n1 = n1s[e];
    int   n2 = n2s[e];
    float we = w[e];
    const float4 v = *(const float4*)(poi + (size_t)n2 * D_FEAT + lane * 4);
    float* dst = x + (size_t)n1 * D_FEAT + lane * 4;
    atomicAdd(dst + 0, we * v.x);
    atomicAdd(dst + 1, we * v.y);
    atomicAdd(dst + 2, we * v.z);
    atomicAdd(dst + 3, we * v.w);
}

// ---- kernel 5: out = x @ W^T + bias  via V_WMMA_F32_16X16X4_F32 ------------
// Block = 256 threads = 8 waves; block owns 16 rows (m0), wave i owns columns
// n0 = 16*i. K = 128 -> 32 chained WMMAs per wave. The 8 waves share the same
// 16 A-rows (WGP$ hit); W (64 KB) stays resident in L2.
//
// ISA fragment layouts (CDNA5 ISA, 32-bit matrices):
//   A 16x4 : lane r (0-15): v0=K0,v1=K1 ; lane 16+r: v0=K2,v1=K3
//            -> a = *(v2f*)&Arow[kk + 2*half]            (8B contiguous)
//   B 4x16 : v0 = rows K0|K2, v1 = rows K1|K3, N = lane&15
//            -> B[k][n] = W[n*128+k]  => b = *(v2f*)&Wrow[kk + 2*half]
//   C 16x16: VGPR j -> row m0 + j + 8*half, col n0 + (lane&15)
__global__ void __launch_bounds__(256)
gemm_wmma_kernel(const float* __restrict__ x, const float* __restrict__ W,
                 const float* __restrict__ bias, float* __restrict__ out, int M) {
    const int wave = threadIdx.x >> 5;   // 0..7 -> N tile
    const int lane = threadIdx.x & 31;
    const int half = lane >> 4;
    const int r    = lane & 15;
    const int m0   = blockIdx.x * 16;
    const int n0   = wave * 16;
    if (m0 >= M) return;

    const float* arow = x + (size_t)(m0 + r) * D_FEAT + 2 * half;
    const float* brow = W + (size_t)(n0 + r) * D_FEAT + 2 * half;

    v8f c = {};
#pragma unroll
    for (int kk = 0; kk < D_FEAT; kk += 4) {
        v2f a = *(const v2f*)(arow + kk);
        v2f b = *(const v2f*)(brow + kk);
        // 8 args: (neg_a, A, neg_b, B, c_mod, C, reuse_a, reuse_b)
        c = __builtin_amdgcn_wmma_f32_16x16x4_f32(false, a, false, b,
                                                  (short)0, c, false, false);
    }

    const float bv = bias[n0 + r];
#pragma unroll
    for (int j = 0; j < 8; ++j) {
        out[(size_t)(m0 + j + 8 * half) * D_FEAT + n0 + r] = c[j] + bv;
    }
}

// ---------------------------------------------------------------------------
extern "C" void kernel_launch(void* const* d_in, const int* in_sizes, int n_in,
                              void* d_out, int out_size, void* d_ws, size_t ws_size,
                              hipStream_t stream) {
    const float* poi  = (const float*)d_in[0];   // [N,128]
    const float* W    = (const float*)d_in[1];   // [128,128]
    const float* bias = (const float*)d_in[2];   // [128]
    const int*   edge = (const int*)d_in[3];     // [2,E]
    const float* dist = (const float*)d_in[4];   // [E]
    float*       out  = (float*)d_out;           // [N,128]

    const int N = in_sizes[0] / D_FEAT;          // 100000
    const int E = in_sizes[4];                   // 1600000
    const int* n1 = edge;                        // row 0
    const int* n2 = edge + E;                    // row 1

    // workspace layout (floats): deg[N] | x[N*128] | w[E]
    float* deg = (float*)d_ws;
    float* x   = deg + N;
    float* w   = x + (size_t)N * D_FEAT;

    // 1) zero deg + x  (N + N*128 floats; both divisible by 4)
    {
        int n4 = (N + N * D_FEAT) / 4;
        zero_f4_kernel<<<(n4 + 255) / 256, 256, 0, stream>>>((float4*)d_ws, n4);
    }
    // 2) degree histogram
    deg_kernel<<<(E + 255) / 256, 256, 0, stream>>>(n1, deg, E);
    // 3) per-edge weights
    weight_kernel<<<(E + 255) / 256, 256, 0, stream>>>(n1, n2, dist, deg, w, E);
    // 4) scatter-add aggregate (one wave32 per edge -> E*32 threads)
    {
        long long threads = (long long)E * 32;
        int blocks = (int)((threads + 255) / 256);
        scatter_kernel<<<blocks, 256, 0, stream>>>(poi, n1, n2, w, x, E);
    }
    // 5) dense GEMM + bias via WMMA (M = 100000 is an exact multiple of 16)
    gemm_wmma_kernel<<<(N + 15) / 16, 256, 0, stream>>>(x, W, bias, out, N);
}